// BatchedEdgeGATLayer_53936199303547
// MI455X (gfx1250) — compile-verified
//
#include <hip/hip_runtime.h>
#include <hip/hip_bf16.h>
#include <stdint.h>

#define Bc 4
#define Nc 10000
#define Ec 160000
#define NODE_DIMc 128
#define EDGE_DIMc 64
#define HIDDENc 256
#define HEADSc 8
#define HEAD_DIMc 32
#define LN_EPSc 1e-5f

// padded LDS row strides (dwords): row*stride mod 64 banks spreads lanes 4 apart
#define A_LDS_STRIDE 132   // for 16x128 node tile
#define E_LDS_STRIDE 68    // for 16x64 edge tile

// packed-B sizes: [16 col tiles][K/4 k-steps][32 lanes][2 floats]
#define WN_PK_ELEMS (16 * 32 * 32 * 2)   // K=128 -> 32768 floats
#define WE_PK_ELEMS (16 * 16 * 32 * 2)   // K=64  -> 16384 floats

typedef __attribute__((ext_vector_type(2))) float v2f;
typedef __attribute__((ext_vector_type(8))) float v8f;

// Order-preserving float<->uint encoding for atomicMax-based segment max.
__device__ __forceinline__ unsigned enc_f32(float x) {
  unsigned u = __float_as_uint(x);
  return (u & 0x80000000u) ? ~u : (u | 0x80000000u);
}
__device__ __forceinline__ float dec_f32(unsigned u) {
  return __uint_as_float((u & 0x80000000u) ? (u ^ 0x80000000u) : ~u);
}

// ---------------------------------------------------------------------------
// 0) init: zero agg + seg_sum, seed seg_max with enc(-inf) = 0x007FFFFF
// ---------------------------------------------------------------------------
__global__ void init_kernel(float* __restrict__ agg, unsigned* __restrict__ smax,
                            float* __restrict__ ssum) {
  size_t i = (size_t)blockIdx.x * blockDim.x + threadIdx.x;
  if (i < (size_t)Bc * Nc * HIDDENc) agg[i] = 0.0f;
  if (i < (size_t)Bc * Nc * HEADSc) {
    smax[i] = 0x007FFFFFu;  // enc(-inf)
    ssum[i] = 0.0f;
  }
}

// ---------------------------------------------------------------------------
// 1) pack weights into WMMA B-fragment order:
//    P[((ct*KS + s)*32 + lane)*2 + {0,1}] = W[(4s + 2*(lane>>4) + {0,1})*256 + ct*16 + lane&15]
//    => inner GEMM loop does ONE global_load_b64 per k-step, immediate offsets.
// ---------------------------------------------------------------------------
__device__ __forceinline__ void pack_one(const float* __restrict__ W,
                                         float* __restrict__ P, int i, int KS) {
  int lane = i & 31;
  int s = (i >> 5) % KS;
  int ct = (i >> 5) / KS;
  int ka = 4 * s + ((lane >> 4) << 1);
  int col = ct * 16 + (lane & 15);
  P[(size_t)i * 2 + 0] = W[(size_t)ka * HIDDENc + col];
  P[(size_t)i * 2 + 1] = W[(size_t)(ka + 1) * HIDDENc + col];
}

__global__ void pack_kernel(const float* __restrict__ Wn,
                            const float* __restrict__ Wr,
                            const float* __restrict__ We,
                            float* __restrict__ WnPk, float* __restrict__ WrPk,
                            float* __restrict__ WePk) {
  int i = blockIdx.x * blockDim.x + threadIdx.x;
  if (i < WN_PK_ELEMS / 2) {
    pack_one(Wn, WnPk, i, 32);
    pack_one(Wr, WrPk, i, 32);
  }
  if (i < WE_PK_ELEMS / 2) pack_one(We, WePk, i, 16);
}

// ---------------------------------------------------------------------------
// 2) watt[k][h] = sum_d W_edge[k, h*32+d] * att_edge[h,d]   (64 x 8)
// ---------------------------------------------------------------------------
__global__ void watt_kernel(const float* __restrict__ We,
                            const float* __restrict__ att_e,
                            float* __restrict__ watt) {
  int i = blockIdx.x * blockDim.x + threadIdx.x;
  if (i >= EDGE_DIMc * HEADSc) return;
  int h = i & 7, k = i >> 3;
  float acc = 0.0f;
  #pragma unroll
  for (int d = 0; d < HEAD_DIMc; ++d)
    acc += We[k * HIDDENc + h * HEAD_DIMc + d] * att_e[h * HEAD_DIMc + d];
  watt[i] = acc;  // layout [k][h]
}

// ---------------------------------------------------------------------------
// 3) node GEMM via V_WMMA_F32_16X16X4_F32 (packed-B):
//    H = node_feat @ W_node, R = node_feat @ W_res  (M=B*N, K=128, N=256)
//    8 waves/block: waves 0-3 -> 16 col tiles of H, waves 4-7 -> 16 of R
//    (weight/base selection is wave-uniform and hoisted out of all loops).
// ---------------------------------------------------------------------------
__global__ void node_gemm_kernel(const float* __restrict__ X,
                                 const float* __restrict__ WnPk,
                                 const float* __restrict__ WrPk,
                                 float* __restrict__ H, float* __restrict__ R) {
  __shared__ float As[16 * A_LDS_STRIDE];   // ~8.4 KB
  const int rowTile = blockIdx.x;           // (B*N)/16 tiles

  // cooperative stage: 16x128 floats = 512 float4, 2 per thread
  {
    const size_t base = (size_t)rowTile * 16 * NODE_DIMc;
    #pragma unroll
    for (int it = 0; it < 2; ++it) {
      int i = threadIdx.x + it * 256;       // 0..511
      int r = i >> 5;                       // row 0..15
      int c = (i & 31) << 2;                // col 0,4,...,124
      *(float4*)&As[r * A_LDS_STRIDE + c] =
          *(const float4*)&X[base + (size_t)r * NODE_DIMc + c];
    }
  }
  __syncthreads();

  const int wave = threadIdx.x >> 5;
  const int lane = threadIdx.x & 31;
  const int l15 = lane & 15;
  const int khalf = (lane >> 4) << 1;       // lanes 16-31 carry K+2 (ISA 7.12.2)

  // wave-uniform selection, hoisted out of all loops
  const float* Wpk = (wave < 4) ? WnPk : WrPk;
  float* O = (wave < 4) ? H : R;
  const int w4 = wave & 3;

  #pragma unroll 1
  for (int t = 0; t < 4; ++t) {
    const int ct = w4 * 4 + t;              // col tile 0..15
    const float* bp = Wpk + (size_t)ct * (32 * 32 * 2) + lane * 2;
    v8f acc = {};
    #pragma unroll
    for (int s = 0; s < 32; ++s) {          // K = 128 = 32 steps of 4
      v2f a = *(const v2f*)&As[l15 * A_LDS_STRIDE + 4 * s + khalf]; // ds_load_b64
      v2f b = *(const v2f*)&bp[s * 64];                             // global_load_b64 (imm off)
      acc = __builtin_amdgcn_wmma_f32_16x16x4_f32(false, a, false, b,
                                                  (short)0, acc, false, false);
    }
    // D layout: VGPR j -> M = j (lanes 0-15) / j+8 (lanes 16-31), N = lane%16
    const size_t mbase = (size_t)rowTile * 16 + ((lane >> 4) ? 8 : 0);
    const int nn = ct * 16 + l15;
    #pragma unroll
    for (int j = 0; j < 8; ++j)
      O[(mbase + j) * HIDDENc + nn] = acc[j];
  }
}

// ---------------------------------------------------------------------------
// 4) per-node attention scores: s_src/s_dst[b,n,h] = <h[b,n,h,:], att>
// ---------------------------------------------------------------------------
__global__ void node_score_kernel(const float* __restrict__ H,
                                  const float* __restrict__ att_s,
                                  const float* __restrict__ att_d,
                                  float* __restrict__ ssrc,
                                  float* __restrict__ sdst) {
  size_t i = (size_t)blockIdx.x * blockDim.x + threadIdx.x;
  if (i >= (size_t)Bc * Nc * HEADSc) return;
  int h = (int)(i & 7);
  size_t bn = i >> 3;
  const float* row = H + bn * HIDDENc + h * HEAD_DIMc;
  float a1 = 0.0f, a2 = 0.0f;
  #pragma unroll
  for (int d = 0; d < HEAD_DIMc; ++d) {
    a1 += row[d] * att_s[h * HEAD_DIMc + d];
    a2 += row[d] * att_d[h * HEAD_DIMc + d];
  }
  ssrc[i] = a1;
  sdst[i] = a2;
}

// ---------------------------------------------------------------------------
// 5) per-edge logits + leaky_relu + segment max (ordered-uint atomicMax)
// ---------------------------------------------------------------------------
__global__ void edge_logits_kernel(const float* __restrict__ EA,
                                   const float* __restrict__ watt,
                                   const float* __restrict__ ssrc,
                                   const float* __restrict__ sdst,
                                   const long long* __restrict__ srcI,
                                   const long long* __restrict__ dstI,
                                   float* __restrict__ logits,
                                   unsigned* __restrict__ smax) {
  size_t i = (size_t)blockIdx.x * blockDim.x + threadIdx.x;
  if (i >= (size_t)Bc * Ec * HEADSc) return;
  int h = (int)(i & 7);
  size_t be = i >> 3;               // b*E + e
  size_t e = be % Ec;
  size_t b = be / Ec;
  const float* a = EA + be * EDGE_DIMc;
  float acc = 0.0f;
  #pragma unroll 8
  for (int k = 0; k < EDGE_DIMc; ++k) acc += a[k] * watt[k * HEADSc + h];
  const long long s = srcI[e], d = dstI[e];
  float lg = ssrc[(b * Nc + (size_t)s) * HEADSc + h] +
             sdst[(b * Nc + (size_t)d) * HEADSc + h] + acc;
  lg = (lg > 0.0f) ? lg : 0.2f * lg;  // leaky_relu(0.2)
  logits[i] = lg;
  atomicMax(&smax[(b * Nc + (size_t)d) * HEADSc + h], enc_f32(lg));
}

// ---------------------------------------------------------------------------
// 6) per-edge exp + segment sum
// ---------------------------------------------------------------------------
__global__ void edge_exp_kernel(float* __restrict__ logits,
                                const unsigned* __restrict__ smax,
                                float* __restrict__ ssum,
                                const long long* __restrict__ dstI) {
  size_t i = (size_t)blockIdx.x * blockDim.x + threadIdx.x;
  if (i >= (size_t)Bc * Ec * HEADSc) return;
  int h = (int)(i & 7);
  size_t be = i >> 3;
  size_t e = be % Ec;
  size_t b = be / Ec;
  const long long d = dstI[e];
  float m = dec_f32(smax[(b * Nc + (size_t)d) * HEADSc + h]);
  float ex = expf(logits[i] - m);
  logits[i] = ex;  // overwrite with exp value
  atomicAdd(&ssum[(b * Nc + (size_t)d) * HEADSc + h], ex);
}

// ---------------------------------------------------------------------------
// 7) fused message kernel (packed-B): recompute e-tile with WMMA
//    (16 edges x 16 cols, K=64), msg = alpha * (h[src] + e), scatter-add
//    into agg (L2-resident). 8 waves/block, 2 col tiles each, serial.
// ---------------------------------------------------------------------------
__global__ void edge_msg_kernel(const float* __restrict__ EA,
                                const float* __restrict__ WePk,
                                const float* __restrict__ H,
                                const float* __restrict__ EX,     // exp values
                                const float* __restrict__ SSUM,
                                const long long* __restrict__ srcI,
                                const long long* __restrict__ dstI,
                                float* __restrict__ AGG) {
  __shared__ float Es[16 * E_LDS_STRIDE];   // ~4.3 KB
  const int bt = blockIdx.x;                 // B * (E/16)
  const int b = bt / (Ec / 16);
  const int et = bt % (Ec / 16);
  const size_t e0 = (size_t)et * 16;
  const float* Arow = EA + ((size_t)b * Ec + e0) * EDGE_DIMc;

  // stage 16x64 floats = 256 float4, one per thread
  {
    int i = threadIdx.x;                    // 0..255
    int r = i >> 4;                         // row 0..15
    int c = (i & 15) << 2;                  // col 0,4,...,60
    *(float4*)&Es[r * E_LDS_STRIDE + c] =
        *(const float4*)&Arow[(size_t)r * EDGE_DIMc + c];
    // prefetch the next edge tile of the HBM stream (global_prefetch_b8)
    if (et + 1 < Ec / 16)
      __builtin_prefetch(Arow + 16 * EDGE_DIMc + (size_t)r * EDGE_DIMc + c, 0, 1);
  }
  __syncthreads();

  const int wave = threadIdx.x >> 5;
  const int lane = threadIdx.x & 31;
  const int l15 = lane & 15;
  const int khalf = (lane >> 4) << 1;

  #pragma unroll 1
  for (int t = 0; t < 2; ++t) {
    const int ct = wave * 2 + t;             // col tile 0..15
    const float* bp = WePk + (size_t)ct * (16 * 32 * 2) + lane * 2;
    v8f acc = {};
    #pragma unroll
    for (int s = 0; s < 16; ++s) {           // K = 64 = 16 steps of 4
      v2f a = *(const v2f*)&Es[l15 * E_LDS_STRIDE + 4 * s + khalf];  // ds_load_b64
      v2f bfrag = *(const v2f*)&bp[s * 64];                          // global_load_b64
      acc = __builtin_amdgcn_wmma_f32_16x16x4_f32(false, a, false, bfrag,
                                                  (short)0, acc, false, false);
    }
    const int nn = ct * 16 + l15;
    const int head = nn >> 5;                // 32 dims per head
    const int mbase = (lane >> 4) ? 8 : 0;
    #pragma unroll
    for (int j = 0; j < 8; ++j) {
      const size_t eidx = e0 + mbase + j;
      const long long s = srcI[eidx], d = dstI[eidx];
      const float ex = EX[((size_t)b * Ec + eidx) * HEADSc + head];
      const float den = SSUM[((size_t)b * Nc + (size_t)d) * HEADSc + head];
      const float alpha = ex / den;
      const float hsrc = H[((size_t)b * Nc + (size_t)s) * HIDDENc + nn];
      atomicAdd(&AGG[((size_t)b * Nc + (size_t)d) * HIDDENc + nn],
                alpha * (hsrc + acc[j]));
    }
  }
}

// ---------------------------------------------------------------------------
// 8) residual + LayerNorm + ELU; one block per (b,n), 256 threads
// ---------------------------------------------------------------------------
__global__ void finalize_kernel(const float* __restrict__ AGG,
                                const float* __restrict__ R,
                                const float* __restrict__ gamma,
                                const float* __restrict__ beta,
                                float* __restrict__ out) {
  __shared__ float red[HIDDENc];
  const size_t bn = blockIdx.x;
  const int tid = threadIdx.x;
  float x = AGG[bn * HIDDENc + tid] + R[bn * HIDDENc + tid];

  red[tid] = x;
  __syncthreads();
  #pragma unroll
  for (int s = HIDDENc / 2; s > 0; s >>= 1) {
    if (tid < s) red[tid] += red[tid + s];
    __syncthreads();
  }
  const float mu = red[0] * (1.0f / HIDDENc);
  __syncthreads();

  const float cx = x - mu;
  red[tid] = cx * cx;
  __syncthreads();
  #pragma unroll
  for (int s = HIDDENc / 2; s > 0; s >>= 1) {
    if (tid < s) red[tid] += red[tid + s];
    __syncthreads();
  }
  const float var = red[0] * (1.0f / HIDDENc);

  float y = cx * rsqrtf(var + LN_EPSc) * gamma[tid] + beta[tid];
  out[bn * HIDDENc + tid] = (y > 0.0f) ? y : (expf(y) - 1.0f);  // ELU
}

// ---------------------------------------------------------------------------
extern "C" void kernel_launch(void* const* d_in, const int* in_sizes, int n_in,
                              void* d_out, int out_size, void* d_ws, size_t ws_size,
                              hipStream_t stream) {
  const float* node_feat = (const float*)d_in[0];
  const long long* eidx  = (const long long*)d_in[1];   // int64 (2,E)
  const float* edge_attr = (const float*)d_in[2];
  const float* W_node    = (const float*)d_in[3];
  const float* W_edge    = (const float*)d_in[4];
  const float* W_res     = (const float*)d_in[5];
  const float* att_src   = (const float*)d_in[6];
  const float* att_dst   = (const float*)d_in[7];
  const float* att_edge  = (const float*)d_in[8];
  const float* ln_gamma  = (const float*)d_in[9];
  const float* ln_beta   = (const float*)d_in[10];
  const long long* srcI = eidx;
  const long long* dstI = eidx + Ec;
  float* out = (float*)d_out;

  // workspace carve-up (floats)
  float* ws = (float*)d_ws;
  float* H    = ws; ws += (size_t)Bc * Nc * HIDDENc;    // 40.96 MB
  float* R    = ws; ws += (size_t)Bc * Nc * HIDDENc;    // 40.96 MB
  float* SSRC = ws; ws += (size_t)Bc * Nc * HEADSc;
  float* SDST = ws; ws += (size_t)Bc * Nc * HEADSc;
  float* WATT = ws; ws += EDGE_DIMc * HEADSc;
  float* LOGB = ws; ws += (size_t)Bc * Ec * HEADSc;     // 20.5 MB (logits->exp)
  unsigned* SMAX = (unsigned*)ws; ws += (size_t)Bc * Nc * HEADSc;
  float* SSUM = ws; ws += (size_t)Bc * Nc * HEADSc;
  float* AGG  = ws; ws += (size_t)Bc * Nc * HIDDENc;    // 40.96 MB
  float* WnPk = ws; ws += WN_PK_ELEMS;                  // packed weights
  float* WrPk = ws; ws += WN_PK_ELEMS;
  float* WePk = ws; ws += WE_PK_ELEMS;

  const int T = 256;
  init_kernel<<<(int)(((size_t)Bc * Nc * HIDDENc + T - 1) / T), T, 0, stream>>>(
      AGG, SMAX, SSUM);
  pack_kernel<<<(WN_PK_ELEMS / 2 + T - 1) / T, T, 0, stream>>>(
      W_node, W_res, W_edge, WnPk, WrPk, WePk);
  watt_kernel<<<(EDGE_DIMc * HEADSc + T - 1) / T, T, 0, stream>>>(
      W_edge, att_edge, WATT);
  node_gemm_kernel<<<(Bc * Nc) / 16, T, 0, stream>>>(
      node_feat, WnPk, WrPk, H, R);
  node_score_kernel<<<(int)(((size_t)Bc * Nc * HEADSc + T - 1) / T), T, 0, stream>>>(
      H, att_src, att_dst, SSRC, SDST);
  edge_logits_kernel<<<(int)(((size_t)Bc * Ec * HEADSc + T - 1) / T), T, 0, stream>>>(
      edge_attr, WATT, SSRC, SDST, srcI, dstI, LOGB, SMAX);
  edge_exp_kernel<<<(int)(((size_t)Bc * Ec * HEADSc + T - 1) / T), T, 0, stream>>>(
      LOGB, SMAX, SSUM, dstI);
  edge_msg_kernel<<<Bc * (Ec / 16), T, 0, stream>>>(
      edge_attr, WePk, H, LOGB, SSUM, srcI, dstI, AGG);
  finalize_kernel<<<Bc * Nc, T, 0, stream>>>(AGG, R, ln_gamma, ln_beta, out);
}